// SpatialTransformer_11879879541403
// MI455X (gfx1250) — compile-verified
//
#include <hip/hip_runtime.h>

// SpatialTransformer / dense_image_warp over B*D=192 images, H=W=224, C=2.
// Memory-bound gather: ~270 MB HBM traffic -> ~12 us floor at 23.3 TB/s.
// Strategy: 1 pixel per lane, float2 (b64) gathers/stores, NT cache policy on
// single-use streams (flow in, out), RT (default) on the reused vol array so
// it stays hot in the 192 MB L2.

typedef __attribute__((ext_vector_type(2))) float v2f;

#ifndef WARP_H
#define WARP_H 224
#define WARP_W 224
#endif

__global__ __launch_bounds__(256) void dense_warp_kernel(
    const float* __restrict__ vol,   // [N, H, W, 2]
    const float* __restrict__ flow,  // [N, H, W, 3] (dy, dx, unused)
    float* __restrict__ out,         // [N, H, W, 2]
    int n_pix)                       // N*H*W
{
    int pix = blockIdx.x * (int)blockDim.x + threadIdx.x;
    if (pix >= n_pix) return;

    const unsigned hw = WARP_H * WARP_W;
    unsigned n   = (unsigned)pix / hw;
    unsigned rem = (unsigned)pix - n * hw;
    unsigned y   = rem / WARP_W;
    unsigned x   = rem - y * WARP_W;

    // Flow is streamed exactly once -> non-temporal loads (TH_NT): don't let
    // the 115 MB flow stream evict the reusable vol lines from L2.
    const float* fp = flow + (size_t)pix * 3;
    float dy = __builtin_nontemporal_load(fp + 0);
    float dx = __builtin_nontemporal_load(fp + 1);

    float qy = (float)y - dy;
    float qx = (float)x - dx;

    // tfa.interpolate_bilinear clamping: floor clipped to [0, size-2],
    // alpha clipped to [0, 1].
    float y0f = fminf(fmaxf(floorf(qy), 0.0f), (float)(WARP_H - 2));
    float x0f = fminf(fmaxf(floorf(qx), 0.0f), (float)(WARP_W - 2));
    float ay  = fminf(fmaxf(qy - y0f, 0.0f), 1.0f);
    float ax  = fminf(fmaxf(qx - x0f, 0.0f), 1.0f);
    int y0 = (int)y0f;
    int x0 = (int)x0f;

    // Gather the 4 corners as float2 (C=2, 8-byte aligned) -> global_load_b64.
    // Default (RT) policy: vol has 4-way reuse across neighboring lanes and
    // fits in L2 (77 MB < 192 MB) once the NT streams stay out of the way.
    size_t base = ((size_t)(n * WARP_H + y0) * WARP_W + (unsigned)x0) * 2u;
    const v2f* r0 = (const v2f*)(vol + base);              // row y0
    const v2f* r1 = (const v2f*)(vol + base + 2u * WARP_W); // row y0+1
    v2f tl = r0[0];
    v2f tr = r0[1];
    v2f bl = r1[0];
    v2f br = r1[1];

    v2f top = tl + ax * (tr - tl);
    v2f bot = bl + ax * (br - bl);
    v2f o   = top + ay * (bot - top);

    // Output written once -> non-temporal store (TH_NT).
    __builtin_nontemporal_store(o, (v2f*)out + pix);
}

extern "C" void kernel_launch(void* const* d_in, const int* in_sizes, int n_in,
                              void* d_out, int out_size, void* d_ws, size_t ws_size,
                              hipStream_t stream) {
    const float* vol  = (const float*)d_in[0];  // B*D*H*W*C floats
    const float* flow = (const float*)d_in[1];  // B*D*H*W*3 floats
    float* out = (float*)d_out;                 // B*D*H*W*C floats

    int n_pix = out_size / 2;                   // C = 2
    const int block = 256;                      // 8 wave32 per block
    int grid = (n_pix + block - 1) / block;
    dense_warp_kernel<<<grid, block, 0, stream>>>(vol, flow, out, n_pix);
}